// GNNFiLMLayer_7481833030223
// MI455X (gfx1250) — compile-verified
//
#include <hip/hip_runtime.h>
#include <hip/hip_bf16.h>

// ---------------------------------------------------------------------------
// GNN FiLM layer for MI455X (gfx1250, wave32).
//   msg   = feat @ W_w                (100000x64)
//   film  = feat @ film_w             (100000x128) -> gamma|beta
//   msg   = relu(gamma*msg + beta)
//   h     = segment_sum(msg[edge_src], edge_dst)   (atomic scatter, BW-bound)
//   out   = LayerNorm(h) * ln_scale + ln_bias
// GEMM uses V_WMMA_F32_16X16X4_F32 (fp32 matrix core path, K=4 per op).
// ---------------------------------------------------------------------------

typedef __attribute__((ext_vector_type(2))) float v2f;
typedef __attribute__((ext_vector_type(8))) float v8f;

#define D_IN   64
#define D_OUT  64
#define NTILES 12   // 192 output cols / 16  (4 msg, 4 gamma, 4 beta tiles)

// One wave (32 threads) computes a 16-row x 192-col tile of P = feat@[Ww|Fw],
// applies FiLM+ReLU register-locally, stores the 16x64 msg tile.
__global__ __launch_bounds__(32)
void gemm_film_kernel(const float* __restrict__ feat,
                      const float* __restrict__ Ww,    // 64 x 64, row-major
                      const float* __restrict__ Fw,    // 64 x 128, row-major
                      float* __restrict__ msg,         // N_SRC x 64
                      int nrows)
{
    const int rowBase = blockIdx.x * 16;
    const int lane    = threadIdx.x;        // 0..31
    const int m       = lane & 15;          // A-matrix row for this lane
    const int khalf   = lane >> 4;          // 0: K={0,1}, 1: K={2,3} within step
    const int col     = lane & 15;          // B/C/D column position

    v8f acc[NTILES];
#pragma unroll
    for (int t = 0; t < NTILES; ++t) {
        v8f z = {};
        acc[t] = z;
    }

    // Clamp row for loads so EXEC stays all-ones (WMMA requirement); mask stores.
    int arow_i = rowBase + m;
    if (arow_i > nrows - 1) arow_i = nrows - 1;
    const float* arow = feat + (long long)arow_i * D_IN;

    for (int kIter = 0; kIter < 16; ++kIter) {
        const int k0 = kIter * 4 + khalf * 2;
        // A operand: 16x4 fp32 tile, lane m holds K=k0,k0+1 in 2 VGPRs.
        v2f a = *(const v2f*)(arow + k0);

#pragma unroll
        for (int nt = 0; nt < NTILES; ++nt) {
            v2f b;
            if (nt < 4) {
                const int n = nt * 16 + col;
                b.x = Ww[(k0 + 0) * D_OUT + n];
                b.y = Ww[(k0 + 1) * D_OUT + n];
            } else {
                const int n = (nt - 4) * 16 + col;
                b.x = Fw[(k0 + 0) * (2 * D_OUT) + n];
                b.y = Fw[(k0 + 1) * (2 * D_OUT) + n];
            }
            // D = A(16x4) * B(4x16) + C  -> v_wmma_f32_16x16x4_f32
            acc[nt] = __builtin_amdgcn_wmma_f32_16x16x4_f32(
                false, a, false, b, (short)0, acc[nt], false, false);
        }
    }

    // FiLM + ReLU: for lane/reg (r), msg tile nt, gamma tile nt+4, beta tile
    // nt+8 all live at the SAME lane/register position. Register-local fuse.
    const int mOff = (lane >> 4) * 8;       // C/D layout: upper lanes hold M=r+8
#pragma unroll
    for (int r = 0; r < 8; ++r) {
        const int mm = rowBase + mOff + r;
        if (mm < nrows) {
#pragma unroll
            for (int nt = 0; nt < 4; ++nt) {
                float v = acc[nt + 4][r] * acc[nt][r] + acc[nt + 8][r];
                v = v > 0.0f ? v : 0.0f;
                msg[(long long)mm * D_OUT + nt * 16 + col] = v;
            }
        }
    }
}

// Edge scatter-add: 64 threads per edge, one float atomic per channel.
// Consecutive threads share an edge -> coalesced msg reads + coalesced atomics.
__global__ __launch_bounds__(256)
void scatter_kernel(const int* __restrict__ esrc,
                    const int* __restrict__ edst,
                    const float* __restrict__ msg,
                    float* __restrict__ h,
                    long long total)   // = E * 64
{
    long long idx = (long long)blockIdx.x * 256 + threadIdx.x;
    if (idx >= total) return;
    const int e = (int)(idx >> 6);
    const int c = (int)(idx & 63);
    const int s = esrc[e];
    const int d = edst[e];
    atomicAdd(&h[(long long)d * D_OUT + c], msg[(long long)s * D_OUT + c]);
}

// LayerNorm over 64 channels: one wave per row, 2 channels per lane,
// wave32 butterfly reduction.
__global__ __launch_bounds__(256)
void ln_kernel(const float* __restrict__ h,
               const float* __restrict__ scale,
               const float* __restrict__ bias,
               float* __restrict__ out,
               int nrows)
{
    const int wave = threadIdx.x >> 5;
    const int lane = threadIdx.x & 31;
    const int row  = blockIdx.x * 8 + wave;
    if (row >= nrows) return;

    const float2* hv = (const float2*)(h + (long long)row * D_OUT);
    float2 x = hv[lane];

    float s  = x.x + x.y;
    float sq = x.x * x.x + x.y * x.y;
#pragma unroll
    for (int mask = 16; mask >= 1; mask >>= 1) {
        s  += __shfl_xor(s,  mask, 32);
        sq += __shfl_xor(sq, mask, 32);
    }
    const float mean = s * (1.0f / 64.0f);
    const float var  = sq * (1.0f / 64.0f) - mean * mean;
    const float inv  = rsqrtf(var + 1e-5f);

    const int c0 = lane * 2;
    float2 o;
    o.x = (x.x - mean) * inv * scale[c0 + 0] + bias[c0 + 0];
    o.y = (x.y - mean) * inv * scale[c0 + 1] + bias[c0 + 1];
    ((float2*)(out + (long long)row * D_OUT))[lane] = o;
}

extern "C" void kernel_launch(void* const* d_in, const int* in_sizes, int n_in,
                              void* d_out, int out_size, void* d_ws, size_t ws_size,
                              hipStream_t stream)
{
    const float* feat = (const float*)d_in[0];
    const int*   esrc = (const int*)  d_in[1];
    const int*   edst = (const int*)  d_in[2];
    const float* Ww   = (const float*)d_in[3];
    const float* Fw   = (const float*)d_in[4];
    const float* lns  = (const float*)d_in[5];
    const float* lnb  = (const float*)d_in[6];
    float*       out  = (float*)d_out;

    const int n_src = in_sizes[0] / D_IN;     // 100000
    const int E     = in_sizes[1];            // 1600000
    const int n_dst = out_size / D_OUT;       // 100000

    float* msg = (float*)d_ws;
    float* h   = (float*)((char*)d_ws + (size_t)n_src * D_OUT * sizeof(float));

    // Deterministic per-call: zero the accumulation buffer (capture-legal).
    hipMemsetAsync(h, 0, (size_t)n_dst * D_OUT * sizeof(float), stream);

    // 1) fused GEMM (fp32 WMMA) + FiLM + ReLU
    const int row_tiles = (n_src + 15) / 16;
    gemm_film_kernel<<<row_tiles, 32, 0, stream>>>(feat, Ww, Fw, msg, n_src);

    // 2) edge scatter-add (HBM/atomic bound — the real cost of this layer)
    const long long total = (long long)E * D_OUT;
    const long long blocks = (total + 255) / 256;
    scatter_kernel<<<(int)blocks, 256, 0, stream>>>(esrc, edst, msg, h, total);

    // 3) LayerNorm
    ln_kernel<<<(n_dst + 7) / 8, 256, 0, stream>>>(h, lns, lnb, out, n_dst);
}